// GAT_180388627283
// MI455X (gfx1250) — compile-verified
//
#include <hip/hip_runtime.h>
#include <hip/hip_bf16.h>

typedef __attribute__((ext_vector_type(2))) float v2f;
typedef __attribute__((ext_vector_type(8))) float v8f;

#define NEG_SLOPE 0.2f
#define EPS_F 1e-16f

// ---------------------------------------------------------------------------
// fp32 WMMA GEMM, strip-mined: C[M,N] = A[M,K] x B[K,N], N = NT*16, K = KK.
// - Whole B (KK x N floats) staged in LDS once per block (CDNA5: 320KB/WGP).
// - One wave32 per 16-row strip of C: per K-step it loads ONE A fragment and
//   issues NT independent V_WMMA_F32_16X16X4_F32 against LDS-resident B.
// - Early-out is wave-uniform and placed after __syncthreads, so EXEC is
//   all-ones at every WMMA (ISA requirement).
// Layouts (CDNA5 ISA, 32-bit): lane L, j=L%16, half=L/16:
//   A vgpr v = A[m0+j][k0+2*half+v]; B vgpr v = B[k0+2*half+v][n0+j];
//   D vgpr v = C[m0+8*half+v][n0+j].
// ---------------------------------------------------------------------------
template <int NT, int KK>
__global__ void gat_wmma_gemm_f32(const float* __restrict__ A,
                                  const float* __restrict__ B,
                                  float* __restrict__ C,
                                  int M) {
    constexpr int N = NT * 16;
    __shared__ float Bs[KK * N];

    // Stage B into LDS (coalesced; all threads participate before the guard).
    for (int i = threadIdx.x; i < KK * N; i += blockDim.x) Bs[i] = B[i];
    __syncthreads();

    const int wavesPerBlock = blockDim.x >> 5;
    const int mTile = blockIdx.x * wavesPerBlock + (threadIdx.x >> 5);
    if (mTile * 16 >= M) return;                 // wave-uniform

    const int lane = threadIdx.x & 31;
    const int j    = lane & 15;
    const int half = lane >> 4;

    const float* Arow = A + (size_t)(mTile * 16 + j) * KK;

    v8f acc[NT];
#pragma unroll
    for (int t = 0; t < NT; ++t) acc[t] = (v8f){};

    for (int k0 = 0; k0 < KK; k0 += 4) {
        const int kk = k0 + half * 2;
        v2f a; a.x = Arow[kk]; a.y = Arow[kk + 1];
        const float* b0 = &Bs[kk * N + j];
        const float* b1 = &Bs[(kk + 1) * N + j];
#pragma unroll
        for (int t = 0; t < NT; ++t) {
            v2f b; b.x = b0[t * 16]; b.y = b1[t * 16];
            acc[t] = __builtin_amdgcn_wmma_f32_16x16x4_f32(
                false, a, false, b, (short)0, acc[t], false, false);
        }
    }

    float* Crow = C + (size_t)(mTile * 16 + half * 8) * N + j;
#pragma unroll
    for (int t = 0; t < NT; ++t)
#pragma unroll
        for (int v = 0; v < 8; ++v)
            Crow[(size_t)v * N + t * 16] = acc[t][v];
}

// ---------------------------------------------------------------------------
// Per-node attention logits: a_src[n,h] = dot(h[n,h,:], att_src[h,:]), same dst.
// ---------------------------------------------------------------------------
__global__ void gat_att_logits(const float* __restrict__ h,
                               const float* __restrict__ att_s,
                               const float* __restrict__ att_d,
                               float* __restrict__ a_s,
                               float* __restrict__ a_d,
                               int NH, int H, int C) {
    int i = blockIdx.x * blockDim.x + threadIdx.x;   // over N*H
    if (i >= NH) return;
    int hd = i % H;
    const float* hp = h + (size_t)i * C;
    const float* ws = att_s + (size_t)hd * C;
    const float* wd = att_d + (size_t)hd * C;
    float ss = 0.f, dd = 0.f;
#pragma unroll 8
    for (int c = 0; c < C; ++c) { float v = hp[c]; ss += v * ws[c]; dd += v * wd[c]; }
    a_s[i] = ss; a_d[i] = dd;
}

// Ordered mapping float -> uint so unsigned atomicMax == float max (no NaNs).
__device__ __forceinline__ unsigned fkey(float f) {
    unsigned u = __float_as_uint(f);
    return (u & 0x80000000u) ? ~u : (u | 0x80000000u);
}
__device__ __forceinline__ float funkey(unsigned k) {
    return (k & 0x80000000u) ? __uint_as_float(k ^ 0x80000000u)
                             : __uint_as_float(~k);
}

// Pass A: raw edge logits + segment-max (ordered-uint atomicMax into maxb).
template <int H>
__global__ void gat_edge_logits(const int* __restrict__ src, const int* __restrict__ dst,
                                int nE, int nLoop,
                                const float* __restrict__ a_s, const float* __restrict__ a_d,
                                float* __restrict__ ebuf, unsigned* __restrict__ maxb) {
    int e = blockIdx.x * blockDim.x + threadIdx.x;
    if (e >= nE + nLoop) return;
    int s, d;
    if (e < nE) { s = src[e]; d = dst[e]; } else { s = d = e - nE; }
#pragma unroll
    for (int hd = 0; hd < H; ++hd) {
        float v = a_s[(size_t)s * H + hd] + a_d[(size_t)d * H + hd];
        v = (v > 0.f) ? v : NEG_SLOPE * v;                 // leaky_relu
        ebuf[(size_t)e * H + hd] = v;
        atomicMax(&maxb[(size_t)d * H + hd], fkey(v));
    }
}

// Pass B: e_exp = exp(e - max[dst]); accumulate softmax denominator.
template <int H>
__global__ void gat_edge_exp(const int* __restrict__ src, const int* __restrict__ dst,
                             int nE, int nLoop,
                             float* __restrict__ ebuf, const unsigned* __restrict__ maxb,
                             float* __restrict__ denom) {
    int e = blockIdx.x * blockDim.x + threadIdx.x;
    if (e >= nE + nLoop) return;
    int d = (e < nE) ? dst[e] : e - nE;
#pragma unroll
    for (int hd = 0; hd < H; ++hd) {
        float m  = funkey(maxb[(size_t)d * H + hd]);
        float ex = expf(ebuf[(size_t)e * H + hd] - m);
        ebuf[(size_t)e * H + hd] = ex;
        atomicAdd(&denom[(size_t)d * H + hd], ex);
    }
}

// Pass C: one thread per (edge, h*C+c): out[dst] += h[src] * alpha.
__global__ void gat_edge_aggr(const int* __restrict__ src, const int* __restrict__ dst,
                              int nE, int nLoop,
                              const float* __restrict__ ebuf, const float* __restrict__ denom,
                              const float* __restrict__ hsrc, float* __restrict__ out,
                              int HC, int C) {
    long long t = (long long)blockIdx.x * blockDim.x + threadIdx.x;
    long long total = (long long)(nE + nLoop) * HC;
    if (t >= total) return;
    int e = (int)(t / HC);
    int idx = (int)(t % HC);
    int hd = idx / C;
    int H = HC / C;
    int s, d;
    if (e < nE) { s = src[e]; d = dst[e]; } else { s = d = e - nE; }
    float alpha = ebuf[(size_t)e * H + hd] / (denom[(size_t)d * H + hd] + EPS_F);
    atomicAdd(&out[(size_t)d * HC + idx], hsrc[(size_t)s * HC + idx] * alpha);
}

// bias + optional ELU, in place.
__global__ void gat_bias_act(float* __restrict__ h, const float* __restrict__ bias,
                             long long total, int ld, int doElu) {
    long long i = (long long)blockIdx.x * blockDim.x + threadIdx.x;
    if (i >= total) return;
    float x = h[i] + bias[i % ld];
    if (doElu) x = (x > 0.f) ? x : (expf(x) - 1.0f);
    h[i] = x;
}

static inline int blocksFor(long long n, int b) { return (int)((n + b - 1) / b); }

extern "C" void kernel_launch(void* const* d_in, const int* in_sizes, int n_in,
                              void* d_out, int out_size, void* d_ws, size_t ws_size,
                              hipStream_t stream) {
    const float* x      = (const float*)d_in[0];
    const int*   ei     = (const int*)  d_in[1];
    const float* W1     = (const float*)d_in[2];
    const float* atts1  = (const float*)d_in[3];
    const float* attd1  = (const float*)d_in[4];
    const float* b1     = (const float*)d_in[5];
    const float* W2     = (const float*)d_in[6];
    const float* atts2  = (const float*)d_in[7];
    const float* attd2  = (const float*)d_in[8];
    const float* b2     = (const float*)d_in[9];

    const int IN = 128, H1 = 4, C1 = 32, HC1 = 128, C2 = 32;
    const int N = in_sizes[0] / IN;          // 50000
    const int E = in_sizes[1] / 2;           // 800000
    const int* src = ei;
    const int* dst = ei + E;

    // ---- workspace layout (~68 MB; resident in the 192 MB L2) ----
    float*    h1     = (float*)d_ws;                         // N*128
    float*    out1   = h1     + (size_t)N * HC1;             // N*128 (-> h2 in place)
    float*    asrc1  = out1   + (size_t)N * HC1;             // N*4
    float*    adst1  = asrc1  + (size_t)N * H1;              // N*4
    float*    denom1 = adst1  + (size_t)N * H1;              // N*4
    unsigned* max1   = (unsigned*)(denom1 + (size_t)N * H1); // N*4
    float*    ebuf   = (float*)(max1 + (size_t)N * H1);      // (E+N)*4
    // layer-2 reuse (strictly after last layer-1 use, stream-ordered):
    float*    h3     = h1;       // N*32
    float*    asrc2  = asrc1;    float* adst2 = adst1;
    float*    denom2 = denom1;   unsigned* max2 = max1;
    float*    e2     = ebuf;     // (E+N)

    float* out = (float*)d_out;  // N*32

    const int BT = 256;                       // 8 wave32 per block
    const int wavesPerBlock = BT / 32;
    const int mTiles = (N + 15) / 16;         // 3125
    const int gemmBlocks = (mTiles + wavesPerBlock - 1) / wavesPerBlock;
    const int nEdgeTot = E + N;

    // =============================== Layer 1 ===============================
    hipMemsetAsync(out1,   0, (size_t)N * HC1 * sizeof(float), stream);
    hipMemsetAsync(max1,   0, (size_t)N * H1  * sizeof(unsigned), stream);
    hipMemsetAsync(denom1, 0, (size_t)N * H1  * sizeof(float), stream);

    // h1 = x @ W1   [N,128] = [N,128]x[128,128]; B staged in 64KB LDS
    gat_wmma_gemm_f32<8, 128><<<gemmBlocks, BT, 0, stream>>>(x, W1, h1, N);

    gat_att_logits<<<blocksFor((long long)N * H1, BT), BT, 0, stream>>>(
        h1, atts1, attd1, asrc1, adst1, N * H1, H1, C1);

    gat_edge_logits<4><<<blocksFor(nEdgeTot, BT), BT, 0, stream>>>(
        src, dst, E, N, asrc1, adst1, ebuf, max1);
    gat_edge_exp<4><<<blocksFor(nEdgeTot, BT), BT, 0, stream>>>(
        src, dst, E, N, ebuf, max1, denom1);
    gat_edge_aggr<<<blocksFor((long long)nEdgeTot * HC1, BT), BT, 0, stream>>>(
        src, dst, E, N, ebuf, denom1, h1, out1, HC1, C1);

    gat_bias_act<<<blocksFor((long long)N * HC1, BT), BT, 0, stream>>>(
        out1, b1, (long long)N * HC1, HC1, /*elu=*/1);       // out1 -> h2

    // =============================== Layer 2 ===============================
    // h3 = h2 @ W2   [N,32] = [N,128]x[128,32]; B staged in 16KB LDS
    gat_wmma_gemm_f32<2, 128><<<gemmBlocks, BT, 0, stream>>>(out1, W2, h3, N);

    gat_att_logits<<<blocksFor((long long)N, BT), BT, 0, stream>>>(
        h3, atts2, attd2, asrc2, adst2, N, 1, C2);

    hipMemsetAsync(max2,   0, (size_t)N * sizeof(unsigned), stream);
    hipMemsetAsync(denom2, 0, (size_t)N * sizeof(float), stream);
    hipMemsetAsync(out,    0, (size_t)N * C2 * sizeof(float), stream);

    gat_edge_logits<1><<<blocksFor(nEdgeTot, BT), BT, 0, stream>>>(
        src, dst, E, N, asrc2, adst2, e2, max2);
    gat_edge_exp<1><<<blocksFor(nEdgeTot, BT), BT, 0, stream>>>(
        src, dst, E, N, e2, max2, denom2);
    gat_edge_aggr<<<blocksFor((long long)nEdgeTot * C2, BT), BT, 0, stream>>>(
        src, dst, E, N, e2, denom2, h3, out, C2, C2);

    gat_bias_act<<<blocksFor((long long)N * C2, BT), BT, 0, stream>>>(
        out, b2, (long long)N * C2, C2, /*elu=*/0);
}